// GML_12068858102100
// MI455X (gfx1250) — compile-verified
//
#include <hip/hip_runtime.h>
#include <hip/hip_bf16.h>
#include <math.h>

// ---------------------------------------------------------------------------
// Problem constants (reference: B=8, N=N1=M1=1024, D=64)
// ---------------------------------------------------------------------------
#define BATCH 8
#define NPTS  1024          // real points per cloud
#define PQ    2048          // padded cloud size (N + N1)
#define DIM   64
#define GAMMA 10.0f
#define MARGIN 0.1f

typedef __attribute__((ext_vector_type(2))) float v2f;
typedef __attribute__((ext_vector_type(8))) float v8f;

// V_WMMA_F32_16X16X4_F32:  D(16x16 f32) = A(16x4 f32) * B(4x16 f32) + C
__device__ __forceinline__ v8f wmma_f32_k4(v2f a, v2f b, v8f c) {
  return __builtin_amdgcn_wmma_f32_16x16x4_f32(false, a, false, b, (short)0, c,
                                               false, false);
}

// ---------------------------------------------------------------------------
// Row L2-normalize: dst row r of batch b comes from src0 (r<1024) or src1.
// One wave per 64-float row; float2 per lane.
// ---------------------------------------------------------------------------
__global__ void k_normalize(const float* __restrict__ src0,
                            const float* __restrict__ src1,
                            float* __restrict__ dst) {
  int row  = blockIdx.x * (blockDim.x >> 5) + (threadIdx.x >> 5); // 0..B*PQ-1
  int lane = threadIdx.x & 31;
  int b = row >> 11;
  int r = row & (PQ - 1);
  const float* s = (r < NPTS)
      ? (src0 + ((size_t)b * NPTS + r) * DIM)
      : (src1 + ((size_t)b * NPTS + (r - NPTS)) * DIM);
  float2 v = ((const float2*)s)[lane];
  float ss = v.x * v.x + v.y * v.y;
  #pragma unroll
  for (int off = 16; off; off >>= 1) ss += __shfl_xor(ss, off);
  float scale = 1.0f / fmaxf(sqrtf(ss), 1e-12f);
  float2 o; o.x = v.x * scale; o.y = v.y * scale;
  ((float2*)(dst + (size_t)row * DIM))[lane] = o;
}

// ---------------------------------------------------------------------------
// sim = exp(GAMMA * Fn1 . Fn2^T) over full [B,2048,2048] via fp32 WMMA with
// 2x2 register blocking (one 32x32 output region per wave, 4 WMMAs/K-step
// sharing 2 A-frags + 2 B-frags -> half the operand traffic per output).
// Stores only the top-left 1024x1024 block into C; accumulates full row sums
// simr[p] (over q) and col sums simc[q] (over p) with wave-reduced atomics.
// ---------------------------------------------------------------------------
__global__ void k_sim(const float* __restrict__ Fn1,
                      const float* __restrict__ Fn2,
                      float* __restrict__ C,
                      float* __restrict__ simr,
                      float* __restrict__ simc) {
  int b    = blockIdx.y;
  int wave = threadIdx.x >> 5;
  int lane = threadIdx.x & 31;
  int t  = blockIdx.x * 8 + wave;      // 0..4095 (64x64 blocks of 32x32)
  int p0 = (t >> 6) * 32;
  int q0 = (t & 63) * 32;
  int mn    = lane & 15;               // A: M index ; B: N index
  int khalf = lane >> 4;               // selects K pair within K=4 step
  const float* A0 = Fn1 + ((size_t)b * PQ + p0 + mn) * DIM;
  const float* A1 = A0 + 16 * DIM;
  const float* B0 = Fn2 + ((size_t)b * PQ + q0 + mn) * DIM;
  const float* B1 = B0 + 16 * DIM;
  v8f acc00 = {}, acc01 = {}, acc10 = {}, acc11 = {};
  #pragma unroll
  for (int k0 = 0; k0 < DIM; k0 += 4) {
    int ko = k0 + 2 * khalf;
    v2f a0 = *(const v2f*)(A0 + ko);
    v2f a1 = *(const v2f*)(A1 + ko);
    v2f b0 = *(const v2f*)(B0 + ko);
    v2f b1 = *(const v2f*)(B1 + ko);
    acc00 = wmma_f32_k4(a0, b0, acc00);
    acc01 = wmma_f32_k4(a0, b1, acc01);
    acc10 = wmma_f32_k4(a1, b0, acc10);
    acc11 = wmma_f32_k4(a1, b1, acc11);
  }
  float s00[8], s01[8], s10[8], s11[8];
  float c0 = 0.0f, c1 = 0.0f;
  #pragma unroll
  for (int r = 0; r < 8; r++) {
    s00[r] = __expf(GAMMA * acc00[r]);
    s01[r] = __expf(GAMMA * acc01[r]);
    s10[r] = __expf(GAMMA * acc10[r]);
    s11[r] = __expf(GAMMA * acc11[r]);
    c0 += s00[r] + s10[r];
    c1 += s01[r] + s11[r];
  }
  // top-left block store (lane holds M = r + 8*khalf, N = mn within tile)
  if (p0 < NPTS && q0 < NPTS) {
    float* Cb = C + ((size_t)b << 20);
    #pragma unroll
    for (int r = 0; r < 8; r++) {
      int m = r + 8 * khalf;
      Cb[(size_t)(p0 + m) * NPTS + q0 + mn]           = s00[r];
      Cb[(size_t)(p0 + m) * NPTS + q0 + 16 + mn]      = s01[r];
      Cb[(size_t)(p0 + 16 + m) * NPTS + q0 + mn]      = s10[r];
      Cb[(size_t)(p0 + 16 + m) * NPTS + q0 + 16 + mn] = s11[r];
    }
  }
  // row sums: reduce across the 16 lanes (over N) of each half
  #pragma unroll
  for (int r = 0; r < 8; r++) {
    float v = s00[r] + s01[r];
    v += __shfl_xor(v, 1); v += __shfl_xor(v, 2);
    v += __shfl_xor(v, 4); v += __shfl_xor(v, 8);
    if (mn == 0)
      atomicAdd(&simr[(size_t)b * PQ + p0 + r + 8 * khalf], v);
    float w = s10[r] + s11[r];
    w += __shfl_xor(w, 1); w += __shfl_xor(w, 2);
    w += __shfl_xor(w, 4); w += __shfl_xor(w, 8);
    if (mn == 0)
      atomicAdd(&simr[(size_t)b * PQ + p0 + 16 + r + 8 * khalf], w);
  }
  // col sums: fold the two lane halves (M groups), one atomic per column
  c0 += __shfl_xor(c0, 16);
  c1 += __shfl_xor(c1, 16);
  if (khalf == 0) {
    atomicAdd(&simc[(size_t)b * PQ + q0 + mn], c0);
    atomicAdd(&simc[(size_t)b * PQ + q0 + 16 + mn], c1);
  }
}

// ---------------------------------------------------------------------------
// Cost epilogue + transpose:  C[p][q] = 1 - s/(simr[p]+simc[q]-s) in place,
// and CT[q][p] = C[p][q] via a 32x33 LDS tile (both sides coalesced).
// Block = 256 threads handles one 32x32 tile.
// ---------------------------------------------------------------------------
__global__ void k_cost_t(float* __restrict__ C, float* __restrict__ CT,
                         const float* __restrict__ simr,
                         const float* __restrict__ simc) {
  __shared__ float tile[32][33];
  int b  = blockIdx.y;
  int p0 = (blockIdx.x >> 5) * 32;
  int q0 = (blockIdx.x & 31) * 32;
  int tx = threadIdx.x & 31;
  int ty = threadIdx.x >> 5;           // 8 rows per pass
  float* Cb  = C  + ((size_t)b << 20);
  float* CTb = CT + ((size_t)b << 20);
  #pragma unroll
  for (int yy = 0; yy < 32; yy += 8) {
    int p = p0 + ty + yy;
    size_t idx = (size_t)p * NPTS + q0 + tx;
    float s = Cb[idx];
    float denom = simr[(size_t)b * PQ + p] + simc[(size_t)b * PQ + q0 + tx] - s;
    float c = 1.0f - s / denom;
    Cb[idx] = c;
    tile[ty + yy][tx] = c;
  }
  __syncthreads();
  #pragma unroll
  for (int yy = 0; yy < 32; yy += 8) {
    int q = q0 + ty + yy;
    CTb[(size_t)q * NPTS + p0 + tx] = tile[tx][ty + yy];
  }
}

// ---------------------------------------------------------------------------
// Hinge: sum relu(MARGIN - x3 . x4^T) via fp32 WMMA, 2x2 register blocking,
// wave-reduced atomicAdd.
// ---------------------------------------------------------------------------
__global__ void k_hinge(const float* __restrict__ X3,
                        const float* __restrict__ X4,
                        float* __restrict__ accum) {
  int b    = blockIdx.y;
  int wave = threadIdx.x >> 5;
  int lane = threadIdx.x & 31;
  int t  = blockIdx.x * 8 + wave;      // 0..1023 (32x32 blocks of 32x32)
  int p0 = (t >> 5) * 32;
  int q0 = (t & 31) * 32;
  int mn    = lane & 15;
  int khalf = lane >> 4;
  const float* A0 = X3 + ((size_t)b * NPTS + p0 + mn) * DIM;
  const float* A1 = A0 + 16 * DIM;
  const float* B0 = X4 + ((size_t)b * NPTS + q0 + mn) * DIM;
  const float* B1 = B0 + 16 * DIM;
  v8f acc00 = {}, acc01 = {}, acc10 = {}, acc11 = {};
  #pragma unroll
  for (int k0 = 0; k0 < DIM; k0 += 4) {
    int ko = k0 + 2 * khalf;
    v2f a0 = *(const v2f*)(A0 + ko);
    v2f a1 = *(const v2f*)(A1 + ko);
    v2f b0 = *(const v2f*)(B0 + ko);
    v2f b1 = *(const v2f*)(B1 + ko);
    acc00 = wmma_f32_k4(a0, b0, acc00);
    acc01 = wmma_f32_k4(a0, b1, acc01);
    acc10 = wmma_f32_k4(a1, b0, acc10);
    acc11 = wmma_f32_k4(a1, b1, acc11);
  }
  float local = 0.0f;
  #pragma unroll
  for (int r = 0; r < 8; r++) {
    local += fmaxf(MARGIN - acc00[r], 0.0f);
    local += fmaxf(MARGIN - acc01[r], 0.0f);
    local += fmaxf(MARGIN - acc10[r], 0.0f);
    local += fmaxf(MARGIN - acc11[r], 0.0f);
  }
  #pragma unroll
  for (int off = 16; off; off >>= 1) local += __shfl_xor(local, off);
  if (lane == 0) atomicAdd(&accum[0], local);
}

// ---------------------------------------------------------------------------
// Softmin along the fast axis of M (used for BOTH directions: pass C for the
// row direction, CT for the column direction):
//   out[b][p] = -eps * LSE_q( h[b][q]*h_scale - M[b][p][q]/eps )
// One wave per row; float4 streams; online (m,s) logsumexp; shfl combine.
// ---------------------------------------------------------------------------
__global__ void k_softmin(const float* __restrict__ M,
                          const float* __restrict__ h,
                          float h_scale, float eps,
                          float* __restrict__ out) {
  int row  = blockIdx.x * (blockDim.x >> 5) + (threadIdx.x >> 5); // 0..8191
  int lane = threadIdx.x & 31;
  int b = row >> 10, p = row & (NPTS - 1);
  const float4* Mrow = (const float4*)(M + ((size_t)b << 20) + (size_t)p * NPTS);
  const float4* hb   = (const float4*)(h + (size_t)b * NPTS);
  float ie = 1.0f / eps;
  float m = -INFINITY, s = 0.0f;
  for (int i = lane; i < NPTS / 4; i += 32) {
    float4 cv = Mrow[i];
    float4 hv = hb[i];
    float x;
    x = hv.x * h_scale - cv.x * ie;
    if (x > m) { s = s * __expf(m - x) + 1.0f; m = x; } else { s += __expf(x - m); }
    x = hv.y * h_scale - cv.y * ie;
    if (x > m) { s = s * __expf(m - x) + 1.0f; m = x; } else { s += __expf(x - m); }
    x = hv.z * h_scale - cv.z * ie;
    if (x > m) { s = s * __expf(m - x) + 1.0f; m = x; } else { s += __expf(x - m); }
    x = hv.w * h_scale - cv.w * ie;
    if (x > m) { s = s * __expf(m - x) + 1.0f; m = x; } else { s += __expf(x - m); }
  }
  #pragma unroll
  for (int off = 16; off; off >>= 1) {
    float mo = __shfl_xor(m, off), so = __shfl_xor(s, off);
    float mx = fmaxf(m, mo);
    s = s * __expf(m - mx) + so * __expf(mo - mx);
    m = mx;
  }
  if (lane == 0) out[row] = -eps * (m + __logf(s));
}

// f = 0.5*(f+ft), g = 0.5*(g+gt)
__global__ void k_avg(float* __restrict__ f, const float* __restrict__ ft,
                      float* __restrict__ g, const float* __restrict__ gt) {
  int i = blockIdx.x * blockDim.x + threadIdx.x;   // 0..8191
  f[i] = 0.5f * (f[i] + ft[i]);
  g[i] = 0.5f * (g[i] + gt[i]);
}

// accum[1] += sum(a) + sum(b)   (a,b each 8192 floats)
__global__ void k_reduce(const float* __restrict__ a,
                         const float* __restrict__ b,
                         float* __restrict__ accum) {
  int i = blockIdx.x * blockDim.x + threadIdx.x;   // 0..8191
  float v = a[i] + b[i];
  #pragma unroll
  for (int off = 16; off; off >>= 1) v += __shfl_xor(v, off);
  __shared__ float w[8];
  int lane = threadIdx.x & 31, wv = threadIdx.x >> 5;
  if (lane == 0) w[wv] = v;
  __syncthreads();
  if (threadIdx.x == 0) {
    float t = 0.0f;
    #pragma unroll
    for (int k = 0; k < 8; k++) t += w[k];
    atomicAdd(&accum[1], t);
  }
}

__global__ void k_finalize(const float* __restrict__ accum,
                           float* __restrict__ out) {
  if (threadIdx.x == 0) {
    out[0] = accum[0];                         // hinge
    out[1] = accum[1] * (1.0f / (float)BATCH); // sinkhorn mean over batch
  }
}

// ---------------------------------------------------------------------------
// Host-side orchestration (graph-capture safe: only async ops on `stream`).
// ---------------------------------------------------------------------------
extern "C" void kernel_launch(void* const* d_in, const int* in_sizes, int n_in,
                              void* d_out, int out_size, void* d_ws, size_t ws_size,
                              hipStream_t stream) {
  const float* x1 = (const float*)d_in[0];
  const float* x2 = (const float*)d_in[1];
  const float* x3 = (const float*)d_in[2];
  const float* x4 = (const float*)d_in[3];

  float* ws   = (float*)d_ws;
  float* Fn1  = ws;                        // 8*2048*64 = 1,048,576
  float* Fn2  = Fn1  + 1048576;            // 1,048,576
  float* C    = Fn2  + 1048576;            // 8M floats (32 MB)
  float* CT   = C    + 8388608;            // 8M floats (32 MB, transposed)
  float* simr = CT   + 8388608;            // 8*2048
  float* simc = simr + BATCH * PQ;         // 8*2048
  float* f    = simc + BATCH * PQ;         // 8*1024
  float* g    = f    + BATCH * NPTS;
  float* ft   = g    + BATCH * NPTS;
  float* gt   = ft   + BATCH * NPTS;
  float* accum= gt   + BATCH * NPTS;       // 2

  // zero simr, simc, f, g (contiguous) and accum
  hipMemsetAsync(simr, 0, (size_t)(2 * BATCH * PQ + 2 * BATCH * NPTS) * sizeof(float), stream);
  hipMemsetAsync(accum, 0, 2 * sizeof(float), stream);

  // normalize padded clouds (8*2048 rows each, 8 rows/block)
  k_normalize<<<BATCH * PQ / 8, 256, 0, stream>>>(x1, x3, Fn1);
  k_normalize<<<BATCH * PQ / 8, 256, 0, stream>>>(x2, x4, Fn2);

  // full-size sim GEMM (fp32 WMMA, 2x2 blocked) + row/col sums + block store
  k_sim<<<dim3(4096 / 8, BATCH), 256, 0, stream>>>(Fn1, Fn2, C, simr, simc);
  // cost epilogue + transposed copy (both L2-resident)
  k_cost_t<<<dim3(1024, BATCH), 256, 0, stream>>>(C, CT, simr, simc);

  // hinge (independent of sinkhorn chain)
  k_hinge<<<dim3(1024 / 8, BATCH), 256, 0, stream>>>(x3, x4, accum);

  // geomloss epsilon schedule: [diam^2, exp(arange(2ln3 -> 2ln0.05, 2ln0.5)), blur^2]
  const float EPS[8] = {9.0f, 9.0f, 2.25f, 0.5625f, 0.140625f,
                        0.03515625f, 0.0087890625f, 0.0025f};

  // init duals: h-term weight 0 (b_log = a_log = 0 on real points)
  k_softmin<<<1024, 256, 0, stream>>>(C,  g, 0.0f, EPS[0], f);
  k_softmin<<<1024, 256, 0, stream>>>(CT, f, 0.0f, EPS[0], g);

  // symmetric Sinkhorn iterations with averaging
  for (int it = 0; it < 8; it++) {
    float eps = EPS[it], ie = 1.0f / eps;
    k_softmin<<<1024, 256, 0, stream>>>(C,  g, ie, eps, ft);
    k_softmin<<<1024, 256, 0, stream>>>(CT, f, ie, eps, gt);
    k_avg<<<BATCH * NPTS / 256, 256, 0, stream>>>(f, ft, g, gt);
  }

  // final extrapolation at eps = blur^2, then loss = sum(f_new) + sum(g_new)
  {
    float eps = EPS[7], ie = 1.0f / eps;
    k_softmin<<<1024, 256, 0, stream>>>(C,  g, ie, eps, ft);
    k_softmin<<<1024, 256, 0, stream>>>(CT, f, ie, eps, gt);
  }
  k_reduce<<<BATCH * NPTS / 256, 256, 0, stream>>>(ft, gt, accum);
  k_finalize<<<1, 32, 0, stream>>>(accum, (float*)d_out);
}